// ABMIL_60644938220089
// MI455X (gfx1250) — compile-verified
//
#include <hip/hip_runtime.h>
#include <math.h>

typedef __attribute__((ext_vector_type(2))) float v2f;
typedef __attribute__((ext_vector_type(4))) float v4f;
typedef __attribute__((ext_vector_type(8))) float v8f;

constexpr int   BB     = 4;
constexpr int   NN     = 16384;
constexpr int   DD     = 1024;
constexpr int   LL     = 4;
constexpr float NEGV   = -10000.0f;

constexpr int NBLK   = 128;          // blocks per bag
constexpr int CHUNK  = NN / NBLK;    // 128 instances per block
constexpr int TPC    = CHUNK / 16;   // 8 tiles of 16 rows per block
constexpr int WGSIZE = 256;          // 8 wave32s
constexpr int NWAVE  = WGSIZE / 32;
constexpr int KSLICE = DD / NWAVE;   // 128 K-columns per wave
constexpr int WS_STRIDE = DD + 4;    // per-(bag,blk) record: m, s, pad, pad, z[1024]

// ---------------------------------------------------------------------------
// Fused kernel: one HBM pass over xs. Per tile of 16 instances:
//   1) scores via v_wmma_f32_16x16x4_f32  (A = xs tile, B = [w1 | w2 | 0...])
//      B-fragment comes from a branchless 3-row LDS table: [w1 | w2 | zeros],
//      each lane holds a fixed row pointer -> uniform ds_load_b64, no exec
//      masking in the WMMA loop.
//   2) online-softmax update of running (max, sum)
//   3) weighted accumulation of z (thread t owns d = 4t..4t+3)
// ---------------------------------------------------------------------------
__global__ __launch_bounds__(WGSIZE)
void abmil_main(const float* __restrict__ xs, const int* __restrict__ valid,
                const float* __restrict__ w1, const float* __restrict__ w2,
                float* __restrict__ ws)
{
    __shared__ float sb[3 * DD];          // row0 = w1, row1 = w2, row2 = zeros
    __shared__ float sred[NWAVE][2][16];
    __shared__ float swt[16];
    __shared__ float sfac;
    __shared__ float sstat[2];            // [0]=running max, [1]=running sum

    const int tid  = threadIdx.x;
    const int wave = tid >> 5;
    const int lane = tid & 31;
    const int half = lane >> 4;           // 0 or 1
    const int ln   = lane & 15;
    const int bag  = blockIdx.x / NBLK;
    const int blk  = blockIdx.x % NBLK;
    const int n0   = blk * CHUNK;
    const float* xbag = xs + (size_t)bag * NN * DD;

    for (int i = tid; i < DD; i += WGSIZE) {
        sb[i]          = w1[i];
        sb[DD + i]     = w2[i];
        sb[2 * DD + i] = 0.0f;
    }
    if (tid == 0) { sstat[0] = -INFINITY; sstat[1] = 0.0f; }
    v4f zacc = {0.0f, 0.0f, 0.0f, 0.0f};
    const int d0 = tid * 4;

    // per-lane fixed B-row: lane 0 -> w1, lane 1 -> w2, all others -> zeros
    const int brow = (ln < 2) ? ln : 2;
    const int kbase = wave * KSLICE + 2 * half;          // shared by A and B
    const float* bbase = &sb[brow * DD + kbase];
    __syncthreads();

    for (int t = 0; t < TPC; ++t) {
        const int r0 = n0 + t * 16;

        // ---- stage 1: gated-attention scores via WMMA --------------------
        // lane reads row (r0+ln), K-slice [wave*128, wave*128+128)
        const float* arow = xbag + (size_t)(r0 + ln) * DD + kbase;
        if (t + 1 < TPC)
            __builtin_prefetch(arow + (size_t)16 * DD, 0, 0);

        v8f acc = {};
        #pragma unroll 8
        for (int k = 0; k < KSLICE; k += 4) {
            v2f af = *(const v2f*)(arow + k);
            v2f bf = *(const v2f*)(bbase + k);
            acc = __builtin_amdgcn_wmma_f32_16x16x4_f32(
                false, af, false, bf, (short)0, acc, false, false);
        }
        // columns N=0 (w1 dot) and N=1 (w2 dot) live in lanes {0,1,16,17}
        if (ln < 2) {
            #pragma unroll
            for (int i = 0; i < 8; ++i)
                sred[wave][ln][half * 8 + i] = acc[i];
        }
        __syncthreads();

        // ---- stage 2: activation + online softmax (wave 0) ---------------
        if (wave == 0) {
            float s1 = 0.0f, s2 = 0.0f;
            #pragma unroll
            for (int w = 0; w < NWAVE; ++w) {
                s1 += sred[w][0][ln];
                s2 += sred[w][1][ln];
            }
            float a = tanhf(s1) * (1.0f / (1.0f + __expf(-s2)));
            const int v = valid[bag * NN + r0 + ln];
            a = v ? a : NEGV;

            float amax = a;                       // reduce within 16-lane group
            for (int off = 8; off > 0; off >>= 1)
                amax = fmaxf(amax, __shfl_xor(amax, off, 32));
            const float m_old = sstat[0];
            const float m_new = fmaxf(m_old, amax);
            const float e = __expf(a - m_new);
            float esum = e;
            for (int off = 8; off > 0; off >>= 1)
                esum += __shfl_xor(esum, off, 32);
            if (lane == 0) {
                const float fac = __expf(m_old - m_new);
                sstat[1] = sstat[1] * fac + esum;
                sstat[0] = m_new;
                sfac = fac;
            }
            if (lane < 16) swt[lane] = e;
        }
        __syncthreads();

        // ---- stage 3: weighted accumulation of z -------------------------
        const float fac = sfac;
        zacc *= fac;
        const float* tbase = xbag + (size_t)r0 * DD + d0;
        #pragma unroll
        for (int m = 0; m < 16; ++m) {
            v4f x4 = *(const v4f*)(tbase + (size_t)m * DD);
            zacc += swt[m] * x4;
        }
        __syncthreads();   // protect sred/swt/sfac for next tile
    }

    // ---- emit per-block record: [m, s, pad, pad, z[1024]] ---------------
    float* rec = ws + (size_t)(bag * NBLK + blk) * WS_STRIDE;
    if (tid == 0) { rec[0] = sstat[0]; rec[1] = sstat[1]; }
    *(v4f*)(rec + 4 + d0) = zacc;
}

// ---------------------------------------------------------------------------
// Combine: merge NBLK partials per bag, normalize, apply wf -> out[B, L]
// ---------------------------------------------------------------------------
__global__ __launch_bounds__(WGSIZE)
void abmil_combine(const float* __restrict__ ws, const float* __restrict__ wf,
                   float* __restrict__ out)
{
    __shared__ float red[WGSIZE];
    const int tid = threadIdx.x;
    const int bag = blockIdx.x;
    const float* base = ws + (size_t)bag * NBLK * WS_STRIDE;

    float m = -INFINITY;
    for (int k = 0; k < NBLK; ++k)
        m = fmaxf(m, base[(size_t)k * WS_STRIDE]);
    float s = 0.0f;
    for (int k = 0; k < NBLK; ++k)
        s += base[(size_t)k * WS_STRIDE + 1] *
             __expf(base[(size_t)k * WS_STRIDE] - m);

    const int d0 = tid * 4;
    v4f z = {0.0f, 0.0f, 0.0f, 0.0f};
    for (int k = 0; k < NBLK; ++k) {
        const float* rec = base + (size_t)k * WS_STRIDE;
        const float  c   = __expf(rec[0] - m);
        v4f zb = *(const v4f*)(rec + 4 + d0);
        z += c * zb;
    }
    z *= (1.0f / s);

    for (int l = 0; l < LL; ++l) {
        v4f w4 = *(const v4f*)(wf + (size_t)l * DD + d0);
        red[tid] = z.x * w4.x + z.y * w4.y + z.z * w4.z + z.w * w4.w;
        __syncthreads();
        for (int off = WGSIZE / 2; off > 0; off >>= 1) {
            if (tid < off) red[tid] += red[tid + off];
            __syncthreads();
        }
        if (tid == 0) out[bag * LL + l] = red[0];
        __syncthreads();
    }
}

extern "C" void kernel_launch(void* const* d_in, const int* in_sizes, int n_in,
                              void* d_out, int out_size, void* d_ws, size_t ws_size,
                              hipStream_t stream)
{
    (void)in_sizes; (void)n_in; (void)out_size; (void)ws_size;
    const float* xs    = (const float*)d_in[0];
    const int*   valid = (const int*)  d_in[1];
    const float* w1    = (const float*)d_in[2];
    const float* w2    = (const float*)d_in[3];
    const float* wf    = (const float*)d_in[4];
    float* out = (float*)d_out;
    float* ws  = (float*)d_ws;   // needs 4*128*1028*4 B ~= 2.1 MB

    abmil_main<<<BB * NBLK, WGSIZE, 0, stream>>>(xs, valid, w1, w2, ws);
    abmil_combine<<<BB, WGSIZE, 0, stream>>>(ws, wf, out);
}